// MaskedDenseLayer_9423158248141
// MI455X (gfx1250) — compile-verified
//
#include <hip/hip_runtime.h>
#include <hip/hip_bf16.h>

// Problem constants (match reference)
#define BS      1024
#define D_IN    512
#define D_OUT   512
#define N_MASKS 8

typedef __attribute__((ext_vector_type(2))) float v2f;
typedef __attribute__((ext_vector_type(8))) float v8f;

// ---------------------------------------------------------------------------
// Pass 1: bucket rows by state.  ws layout: counts[8] | offs[8] | perm[1024]
// Single block of 1024 threads; all accounting in LDS, written out once.
// Within-bucket order is atomic-nondeterministic but the final output is
// order-invariant (each row is written exactly once).
// ---------------------------------------------------------------------------
__global__ void __launch_bounds__(BS)
made_bucket_kernel(const int* __restrict__ state,
                   int* __restrict__ counts,
                   int* __restrict__ offs,
                   int* __restrict__ perm) {
    __shared__ int s_cnt[N_MASKS];
    __shared__ int s_off[N_MASKS];
    __shared__ int s_cur[N_MASKS];

    const int t = threadIdx.x;
    if (t < N_MASKS) s_cnt[t] = 0;
    __syncthreads();

    const int s = state[t];
    atomicAdd(&s_cnt[s], 1);
    __syncthreads();

    if (t == 0) {
        int acc = 0;
        #pragma unroll
        for (int i = 0; i < N_MASKS; ++i) {
            s_off[i] = acc;
            acc += s_cnt[i];
            s_cur[i] = 0;
        }
    }
    __syncthreads();

    const int pos = atomicAdd(&s_cur[s], 1);
    perm[s_off[s] + pos] = t;

    if (t < N_MASKS) {
        counts[t] = s_cnt[t];
        offs[t]   = s_off[t];
    }
}

// ---------------------------------------------------------------------------
// Pass 2: per-bucket masked GEMM via V_WMMA_F32_16X16X4_F32 (fp32 in/accum).
//
// grid.x = N_MASKS * 16 chunk slots (64 rows each), grid.y = D_OUT/64.
// Block = 128 threads = 4 waves; wave w owns the 16-col subtile at
// cols [blockIdx.y*64 + w*16, +16) and accumulates FOUR 16-row tiles
// simultaneously, so each masked-B fragment load feeds 4 WMMAs (4x less
// kernel/mask L2 traffic than one-tile-per-wave).
//
// WMMA f32 16x16x4 fragment layout (ISA 7.12.2):
//   A (16x4):  v0 = {K=0 | K=2}, v1 = {K=1 | K=3}, M = lane%16
//   B (4x16):  v0 = {K=0 | K=2}, v1 = {K=1 | K=3}, N = lane%16
//   C/D (16x16, 8 VGPRs): VGPR g -> M = g + 8*(lane>=16), N = lane%16
// ---------------------------------------------------------------------------
__global__ void __launch_bounds__(128)
made_gemm_kernel(const float* __restrict__ x,
                 const float* __restrict__ kern,
                 const float* __restrict__ masks,
                 const int*   __restrict__ counts,
                 const int*   __restrict__ offs,
                 const int*   __restrict__ perm,
                 float*       __restrict__ out) {
    const int n   = blockIdx.x >> 4;       // mask id
    const int ch  = blockIdx.x & 15;       // 64-row chunk within bucket
    const int cnt = counts[n];
    const int r0  = ch * 64;
    if (r0 >= cnt) return;                 // uniform: wave exits before WMMA

    const int base = offs[n];
    const int wave = threadIdx.x >> 5;
    const int lane = threadIdx.x & 31;
    const int half = lane >> 4;            // 0: K pair {0,1}; 1: K pair {2,3}
    const int l15  = lane & 15;
    const int col  = blockIdx.y * 64 + wave * 16 + l15;

    // Gather rows for the 4 A-fragment tiles (clamped for partial tails; the
    // duplicated rows produce valid-but-unused accum that is never stored).
    const float* __restrict__ xrow[4];
    #pragma unroll
    for (int t = 0; t < 4; ++t) {
        int rA = r0 + t * 16 + l15;
        if (rA >= cnt) rA = cnt - 1;
        xrow[t] = x + (long)perm[base + rA] * D_IN;
    }

    const float* __restrict__ kcol = kern  + col;
    const float* __restrict__ mcol = masks + (long)n * (D_IN * (long)D_OUT) + col;

    v8f acc[4] = {{}, {}, {}, {}};

    #pragma unroll 2
    for (int k = 0; k < D_IN; k += 4) {
        const int kk = k + 2 * half;

        if ((k & 63) == 0 && (k + 64) < D_IN) {
            __builtin_prefetch(mcol + (long)(kk + 64) * D_OUT, 0, 1);
            __builtin_prefetch(kcol + (long)(kk + 64) * D_OUT, 0, 1);
        }

        // One masked-B fragment, reused by 4 WMMAs.
        v2f b;
        b.x = kcol[(long)kk * D_OUT]       * mcol[(long)kk * D_OUT];
        b.y = kcol[(long)(kk + 1) * D_OUT] * mcol[(long)(kk + 1) * D_OUT];

        #pragma unroll
        for (int t = 0; t < 4; ++t) {
            v2f a;
            a.x = xrow[t][kk];
            a.y = xrow[t][kk + 1];
            // 8 args: (neg_a, A, neg_b, B, c_mod, C, reuse_a, reuse_b)
            acc[t] = __builtin_amdgcn_wmma_f32_16x16x4_f32(
                false, a, false, b, (short)0, acc[t], false, false);
        }
    }

    // Scatter-store with fused ReLU.  D VGPR g -> row M = g + 8*half.
    float* __restrict__ ocol = out + col;
    #pragma unroll
    for (int t = 0; t < 4; ++t) {
        #pragma unroll
        for (int g = 0; g < 8; ++g) {
            const int r = r0 + t * 16 + g + 8 * half;
            if (r < cnt) {
                const int row = perm[base + r];
                const float v = acc[t][g];
                ocol[(long)row * D_OUT] = v > 0.0f ? v : 0.0f;
            }
        }
    }
}

// ---------------------------------------------------------------------------
// Host entry
// ---------------------------------------------------------------------------
extern "C" void kernel_launch(void* const* d_in, const int* in_sizes, int n_in,
                              void* d_out, int out_size, void* d_ws, size_t ws_size,
                              hipStream_t stream) {
    const float* x     = (const float*)d_in[0];   // [BS, D_IN]
    const int*   state = (const int*)  d_in[1];   // [BS]
    const float* kern  = (const float*)d_in[2];   // [D_IN, D_OUT]
    const float* masks = (const float*)d_in[3];   // [N_MASKS, D_IN, D_OUT]
    float*       out   = (float*)d_out;           // [BS, D_OUT]

    int* ws     = (int*)d_ws;
    int* counts = ws;                 // 8 ints
    int* offs   = ws + N_MASKS;       // 8 ints
    int* perm   = ws + 2 * N_MASKS;   // 1024 ints

    made_bucket_kernel<<<1, BS, 0, stream>>>(state, counts, offs, perm);

    dim3 grid(N_MASKS * (BS / 64), D_OUT / 64);
    made_gemm_kernel<<<grid, 128, 0, stream>>>(x, kern, masks, counts, offs, perm, out);
}